// _VirtualModule_17514876634087
// MI455X (gfx1250) — compile-verified
//
#include <hip/hip_runtime.h>

// ---------------------------------------------------------------------------
// Scatter-interpolate update:
//   out = state;  out[b, idx[b,s], :] = (1-p[b,s])*state[b, idx[b,s], :]
//                                       + p[b,s]*substate[b, s, :]
// B=32, N=2048, S=64, D=1024 (f32).  Pure streaming problem: ~520 MB of HBM
// traffic, ~22 us at 23.3 TB/s.  No matrix math -> no WMMA; we instead use
// the CDNA5 async global<->LDS data-mover path for the pure-copy rows and
// 128-bit non-temporal vector ops everywhere else.
// ---------------------------------------------------------------------------

typedef float v4f __attribute__((ext_vector_type(4)));
typedef int   v4i __attribute__((ext_vector_type(4)));

#define AS1 __attribute__((address_space(1)))
#define AS3 __attribute__((address_space(3)))

#if defined(__has_builtin)
#if __has_builtin(__builtin_amdgcn_global_load_async_to_lds_b128) &&           \
    __has_builtin(__builtin_amdgcn_global_store_async_from_lds_b128) &&        \
    __has_builtin(__builtin_amdgcn_s_wait_asynccnt)
#define USE_ASYNC_LDS 1
#endif
#endif
#ifndef USE_ASYNC_LDS
#define USE_ASYNC_LDS 0
#endif

static constexpr int kB = 32;
static constexpr int kN = 2048;
static constexpr int kS = 64;
static constexpr int kD = 1024;

__global__ __launch_bounds__(256)
void scatter_interp_kernel(const float* __restrict__ state,
                           const float* __restrict__ substate,
                           const int*   __restrict__ sel_idx,
                           const float* __restrict__ sel_p,
                           float* __restrict__ out)
{
    __shared__ int   s_sel;          // last s with idx[b,s]==n, or -1
    __shared__ float lds_row[kD];    // 4 KB staging for the async copy path

    const int tid = threadIdx.x;     // 0..255, 8 wave32s
    const int row = blockIdx.x;      // row = b*kN + n
    const int b   = row >> 11;       // kN == 2048
    const int n   = row & (kN - 1);

    if (tid == 0) s_sel = -1;
    __syncthreads();
    // Duplicate indices: JAX sequential .at[].set semantics => LAST match wins.
    if (tid < kS) {
        if (sel_idx[b * kS + tid] == n) {
            atomicMax(&s_sel, tid);
        }
    }
    __syncthreads();

    // Each thread owns one 16-byte (float4) slice of this 4 KB row.
    const size_t off = (size_t)row * kD + (size_t)tid * 4;
    const int s = s_sel;             // block-uniform

    if (s >= 0) {
        // Selected row: gather index == scatter index == this row, so the
        // "gathered" data is exactly our row.  Read once, interpolate, write.
        const float p = sel_p[b * kS + s];
        const float q = 1.0f - p;
        v4f g = __builtin_nontemporal_load((const v4f*)(state + off));
        const size_t soff = ((size_t)b * kS + s) * (size_t)kD + (size_t)tid * 4;
        v4f u = *(const v4f*)(substate + soff);
        v4f r;
        r.x = q * g.x + p * u.x;
        r.y = q * g.y + p * u.y;
        r.z = q * g.z + p * u.z;
        r.w = q * g.w + p * u.w;
        __builtin_nontemporal_store(r, (v4f*)(out + off));
    } else {
        // Pure copy row (~97% of rows): route through the CDNA5 async
        // data mover (memory -> LDS -> memory, no VGPR round trip).
#if USE_ASYNC_LDS
        float* l = &lds_row[tid * 4];
        __builtin_amdgcn_global_load_async_to_lds_b128(
            (AS1 v4i*)(state + off), (AS3 v4i*)l, 0, 0);
        // Each lane stages only its own 16 B; draining this wave's ASYNCcnt
        // is sufficient before storing back out (loads complete in order).
        __builtin_amdgcn_s_wait_asynccnt(0);
        __builtin_amdgcn_global_store_async_from_lds_b128(
            (AS1 v4i*)(out + off), (AS3 v4i*)l, 0, 0);
        // S_ENDPGM performs an implicit wait-idle, so the outstanding async
        // store is safe.
#else
        v4f g = __builtin_nontemporal_load((const v4f*)(state + off));
        __builtin_nontemporal_store(g, (v4f*)(out + off));
#endif
    }
}

extern "C" void kernel_launch(void* const* d_in, const int* in_sizes, int n_in,
                              void* d_out, int out_size, void* d_ws, size_t ws_size,
                              hipStream_t stream)
{
    (void)in_sizes; (void)n_in; (void)out_size; (void)d_ws; (void)ws_size;
    const float* state    = (const float*)d_in[0];
    const float* substate = (const float*)d_in[1];
    const int*   sel_idx  = (const int*)  d_in[2];
    const float* sel_p    = (const float*)d_in[3];
    float*       out      = (float*)d_out;

    dim3 grid(kB * kN);   // one block per output row (65536 blocks)
    dim3 block(256);      // 8 wave32s; each thread moves one float4
    scatter_interp_kernel<<<grid, block, 0, stream>>>(state, substate, sel_idx,
                                                      sel_p, out);
}